// Dist_61881888800851
// MI455X (gfx1250) — compile-verified
//
#include <hip/hip_runtime.h>

typedef float v2f __attribute__((ext_vector_type(2)));
typedef float v8f __attribute__((ext_vector_type(8)));
typedef unsigned long long u64;

#define GW 128
#define GH 96
#define NPIX (GW * GH)
#define VTH 0.001f

__global__ void init_kernel(int* cnt) {
    if (threadIdx.x == 0) *cnt = 0;
}

__global__ void compact_kernel(const float* __restrict__ S, int* __restrict__ cnt,
                               float4* __restrict__ feat) {
    int i = blockIdx.x * blockDim.x + threadIdx.x;
    if (i >= NPIX) return;
    float s = S[i];
    if (s > VTH) {
        int p = atomicAdd(cnt, 1);
        float cx = (float)(i % GW);
        float cy = (float)(i / GW);
        // .w carries the linear index BIT-CAST (not converted): shuffles move bits.
        feat[p] = make_float4(cx, cy, cx * cx + cy * cy, __int_as_float(i));
    }
}

__global__ void pad_kernel(const int* __restrict__ cnt, float4* __restrict__ feat) {
    int c = *cnt;
    feat[c + threadIdx.x] = make_float4(0.0f, 0.0f, 1.0e9f, __int_as_float(-1));
}

// Branch-free insert of u64 key = (f32_bits(d2) << 32) | idx into sorted 4-list.
// Single u64 compare == lexicographic (d2 asc, idx asc); lowers to v_min_u64/v_max_u64.
__device__ __forceinline__ void insK(u64 key, u64& k0, u64& k1, u64& k2, u64& k3) {
    u64 t; bool lt;
    lt = key < k0; t = lt ? k0 : key; k0 = lt ? key : k0; key = t;
    lt = key < k1; t = lt ? k1 : key; k1 = lt ? key : k1; key = t;
    lt = key < k2; t = lt ? k2 : key; k2 = lt ? key : k2; key = t;
    lt = key < k3;                    k3 = lt ? key : k3;
}

__global__ __launch_bounds__(256)
void knn_wmma_kernel(const float* __restrict__ S, const int* __restrict__ cnt,
                     const float4* __restrict__ feat, float* __restrict__ outF) {
    const int lane  = threadIdx.x & 31;
    const int wave  = blockIdx.x * (blockDim.x >> 5) + (threadIdx.x >> 5);
    const int qBase = wave * 16;
    const int col   = lane & 15;
    const bool hi   = lane >= 16;
    const int q     = qBase + col;
    const float qx  = (float)(q % GW);
    const float qy  = (float)(q / GW);

    // B operand: VGPR0 = rows {0,2}, VGPR1 = rows {1,3} of the 4x16 query matrix
    v2f bmat;
    bmat.x = hi ? 1.0f : (-2.0f * qx);
    bmat.y = hi ? (qx * qx + qy * qy) : (-2.0f * qy);

    const int M     = *cnt;
    const int tiles = (M + 15) >> 4;
    const int off   = hi ? 8 : 0;

    const u64 KINIT = ((u64)__float_as_uint(3.0e38f) << 32) | 0x7fffffffull;
    // two independent top-4 lists -> two independent min/max dependency chains
    u64 a0 = KINIT, a1 = KINIT, a2 = KINIT, a3 = KINIT;
    u64 b0 = KINIT, b1 = KINIT, b2 = KINIT, b3 = KINIT;

    for (int t = 0; t < tiles; ++t) {
        const int base = t << 4;
        float4 f = feat[base + col];
        // A operand: lanes 0-15 carry K=0,1 = (cx, cy); lanes 16-31 carry K=2,3 = (cx^2+cy^2, 1)
        v2f amat;
        amat.x = hi ? f.z : f.x;
        amat.y = hi ? 1.0f : f.y;
        v8f c = {0.f, 0.f, 0.f, 0.f, 0.f, 0.f, 0.f, 0.f};
        c = __builtin_amdgcn_wmma_f32_16x16x4_f32(false, amat, false, bmat,
                                                  (short)0, c, false, false);
        float wv = f.w;  // lane L holds idx bits of candidate base + (L & 15)
#pragma unroll
        for (int v = 0; v < 4; ++v) {
            unsigned ci = __float_as_uint(__shfl(wv, v + off, 32));
            u64 key = ((u64)__float_as_uint(c[v]) << 32) | ci;
            insK(key, a0, a1, a2, a3);
        }
#pragma unroll
        for (int v = 4; v < 8; ++v) {
            unsigned ci = __float_as_uint(__shfl(wv, v + off, 32));
            u64 key = ((u64)__float_as_uint(c[v]) << 32) | ci;
            insK(key, b0, b1, b2, b3);
        }
    }

    // fold list B into list A
    insK(b0, a0, a1, a2, a3);
    insK(b1, a0, a1, a2, a3);
    insK(b2, a0, a1, a2, a3);
    insK(b3, a0, a1, a2, a3);

    // merge lane q with lane q+16 (disjoint candidate rows)
    u64 p0 = __shfl_down(a0, 16, 32), p1 = __shfl_down(a1, 16, 32);
    u64 p2 = __shfl_down(a2, 16, 32), p3 = __shfl_down(a3, 16, 32);

    if (!hi) {
        insK(p0, a0, a1, a2, a3);
        insK(p1, a0, a1, a2, a3);
        insK(p2, a0, a1, a2, a3);
        insK(p3, a0, a1, a2, a3);

        float d0 = __uint_as_float((unsigned)(a0 >> 32));
        float d1 = __uint_as_float((unsigned)(a1 >> 32));
        float d2 = __uint_as_float((unsigned)(a2 >> 32));
        float d3 = __uint_as_float((unsigned)(a3 >> 32));
        int   i0 = (int)(unsigned)a0, i1 = (int)(unsigned)a1;
        int   i2 = (int)(unsigned)a2, i3 = (int)(unsigned)a3;

        // exact top_k semantics when < 4 valid pixels exist: remaining picks are
        // the lowest-index invalid pixels, ascending. (never triggers at ~5% density)
        if (M < 4) {
            int j = 0;
            if (d0 >= 1.0e8f) { while (j < NPIX && S[j] > VTH) ++j; i0 = j++; }
            if (d1 >= 1.0e8f) { while (j < NPIX && S[j] > VTH) ++j; i1 = j++; }
            if (d2 >= 1.0e8f) { while (j < NPIX && S[j] > VTH) ++j; i2 = j++; }
            if (d3 >= 1.0e8f) { while (j < NPIX && S[j] > VTH) ++j; i3 = j++; }
        }

        int* outI = (int*)outF;
#pragma unroll
        for (int k = 0; k < 4; ++k) {
            int id = (k == 0) ? i0 : (k == 1) ? i1 : (k == 2) ? i2 : i3;
            float fx = (float)(id % GW);
            float fy = (float)(id / GW);
            outF[k * NPIX + q]            = fx;  // IPCnum[0][0][k][:]
            outF[4 * NPIX + k * NPIX + q] = fy;  // IPCnum[0][1][k][:]
            outI[8 * NPIX + k * NPIX + q] = id;  // args[0][k][:]
        }
    }
}

extern "C" void kernel_launch(void* const* d_in, const int* in_sizes, int n_in,
                              void* d_out, int out_size, void* d_ws, size_t ws_size,
                              hipStream_t stream) {
    (void)in_sizes; (void)n_in; (void)out_size; (void)ws_size;
    const float* S = (const float*)d_in[0];
    int*    cnt  = (int*)d_ws;
    float4* feat = (float4*)((char*)d_ws + 16);
    float*  outF = (float*)d_out;

    init_kernel<<<1, 32, 0, stream>>>(cnt);
    compact_kernel<<<NPIX / 256, 256, 0, stream>>>(S, cnt, feat);
    pad_kernel<<<1, 16, 0, stream>>>(cnt, feat);
    // 768 waves total, 16 queries per wave, 8 waves per 256-thread block
    knn_wmma_kernel<<<96, 256, 0, stream>>>(S, cnt, feat, outF);
}